// SlicedVectorQuantize4_3272765079613
// MI455X (gfx1250) — compile-verified
//
#include <hip/hip_runtime.h>
#include <hip/hip_bf16.h>
#include <stdint.h>

// Sliced 4-way vector quantization for MI455X (gfx1250).
//  - distance GEMM on the matrix pipe: V_WMMA_F32_16X16X4_F32 (fp32 needed:
//    argmin margins ~1e-3 vs |x|~11; bf16/f16 would corrupt code selection)
//  - codebook chunks staged global->LDS with GLOBAL_LOAD_ASYNC_TO_LDS_B128
//    (ASYNCcnt), double-buffered; B fragments served from LDS (ds_load_b64,
//    row pad 132 floats -> conflict-free 64-bank access)
//  - x read from HBM exactly once (A fragments stay resident in VGPRs)

typedef float v2f __attribute__((ext_vector_type(2)));
typedef float v8f __attribute__((ext_vector_type(8)));
typedef __attribute__((address_space(3))) float lds_f32;

#define T_DIM   4096
#define C_DIM   512
#define B_DIM   8
#define SUB     128
#define KCODES  512
#define N_ROWS  (B_DIM * T_DIM)          // 32768 rows per sub
#define N_ELEM  (B_DIM * C_DIM * T_DIM)  // 16777216
#define BETA    0.25f
#define WPAD    132                      // padded LDS row (floats)

static __device__ __forceinline__ uint32_t lds_addr(const void* p) {
    // generic -> LDS addrspace cast; AS(3) pointers are raw 32-bit LDS offsets
    return (uint32_t)(uintptr_t)(lds_f32*)(p);
}

// Cooperatively DMA one 16-code x 128-dim codebook chunk into LDS.
// 256 threads x 2 transfers x 16B = 8192B (pad gaps between rows untouched).
static __device__ __forceinline__ void async_stage_chunk(
    const float* __restrict__ ws, int j0, float (*buf)[WPAD], int tid) {
    #pragma unroll
    for (int u = 0; u < 2; ++u) {
        const int e   = (tid + u * 256) * 4;   // flat float index in chunk
        const int row = e >> 7;                // code within chunk
        const int col = e & 127;               // dim
        const uint32_t lds = lds_addr(&buf[row][col]);
        const uint64_t g   = (uint64_t)(uintptr_t)(ws + (size_t)(j0 + row) * SUB + col);
        asm volatile("global_load_async_to_lds_b128 %0, %1, off"
                     :: "v"(lds), "v"(g) : "memory");
    }
}

__global__ __launch_bounds__(256) void vq_main_kernel(
    const float* __restrict__ x,
    const float* __restrict__ w1, const float* __restrict__ w2,
    const float* __restrict__ w3, const float* __restrict__ w4,
    float* __restrict__ out,      // quant, [8,512,4096] float
    float* __restrict__ hist,     // [4*512] code counts (d_ws)
    float* __restrict__ lossAcc)  // [1] sum of squared error (d_ws)
{
    __shared__ float wbuf[2][16][WPAD];  // double-buffered codebook chunk
    __shared__ int   sidx[8 * 16];       // per-wave argmin indices

    const int tid  = threadIdx.x;
    const int lane = tid & 31;
    const int wave = tid >> 5;
    const int m    = lane & 15;        // row within tile / code col within chunk
    const int hi   = lane >> 4;        // K-half selector of the wmma layout

    const int tile = blockIdx.x * 8 + wave;      // 8192 tiles = 4 subs * 2048
    const int s    = tile >> 11;                  // sub id (uniform per block)
    const int rowT = tile & 2047;
    const int n0   = rowT << 4;                   // first of 16 rows (n = b*T + t)
    const int b    = n0 >> 12;
    const int t0   = n0 & (T_DIM - 1);

    const float* ws = (s == 0) ? w1 : (s == 1) ? w2 : (s == 2) ? w3 : w4;

    // Base element index of x[b, c = s*128 + c_rel, t = t0 + m]
    const size_t xbase = ((size_t)(b * C_DIM + s * SUB)) * T_DIM + (size_t)t0 + m;

    // ---- Load the 16x128 A tile once, directly in WMMA f32 16x16x4 A layout.
    v2f af[32];
    #pragma unroll
    for (int kk = 0; kk < 32; ++kk) {
        const size_t c0 = (size_t)(kk * 4 + 2 * hi) * T_DIM;
        af[kk].x = x[xbase + c0];
        af[kk].y = x[xbase + c0 + T_DIM];
    }

    float bestd[8];
    int   bestj[8];
    #pragma unroll
    for (int r = 0; r < 8; ++r) { bestd[r] = __builtin_inff(); bestj[r] = 0; }

    // Prologue: stage chunk 0.
    async_stage_chunk(ws, 0, wbuf[0], tid);
    asm volatile("s_wait_asynccnt 0" ::: "memory");
    __syncthreads();

    // ---- 32 chunks of 16 codes; DMA of chunk jc+1 overlaps compute of jc.
    for (int jc = 0; jc < 32; ++jc) {
        const int cur = jc & 1;
        if (jc + 1 < 32)
            async_stage_chunk(ws, (jc + 1) << 4, wbuf[cur ^ 1], tid);

        const int j = (jc << 4) + m;   // this lane's code column
        v8f acc = {0.f, 0.f, 0.f, 0.f, 0.f, 0.f, 0.f, 0.f};
        float wsq = 0.f;
        #pragma unroll
        for (int kk = 0; kk < 32; ++kk) {
            // B fragment from LDS: conflict-free b64 (row pad = 132 floats)
            const v2f bw = *(const v2f*)(&wbuf[cur][m][kk * 4 + 2 * hi]);
            wsq += bw.x * bw.x + bw.y * bw.y;
            acc = __builtin_amdgcn_wmma_f32_16x16x4_f32(
                false, af[kk], false, bw, (short)0, acc, false, false);
        }
        wsq += __shfl_xor(wsq, 16, 32);   // fold K-halves -> full |w_j|^2

        // score = |w_j|^2 - 2*dot ; |x_m|^2 is row-constant, drop it for argmin
        #pragma unroll
        for (int r = 0; r < 8; ++r) {
            const float d = __builtin_fmaf(-2.0f, acc[r], wsq);
            if (d < bestd[r]) { bestd[r] = d; bestj[r] = j; }
        }

        asm volatile("s_wait_asynccnt 0" ::: "memory");
        __syncthreads();   // chunk jc+1 visible, buf[cur] free for reuse
    }

    // ---- Argmin across the 16 code-columns (stays within each 16-lane half).
    #pragma unroll
    for (int r = 0; r < 8; ++r) {
        float d = bestd[r]; int j = bestj[r];
        #pragma unroll
        for (int mask = 1; mask < 16; mask <<= 1) {
            const float od = __shfl_xor(d, mask, 32);
            const int   oj = __shfl_xor(j, mask, 32);
            if (od < d || (od == d && oj < j)) { d = od; j = oj; }
        }
        bestd[r] = d; bestj[r] = j;
    }

    // Rows M = r + 8*hi. Publish indices + histogram (lanes 0 and 16).
    if (m == 0) {
        #pragma unroll
        for (int r = 0; r < 8; ++r) {
            sidx[wave * 16 + r + 8 * hi] = bestj[r];
            atomicAdd(&hist[s * KCODES + bestj[r]], 1.0f);
        }
    }
    __builtin_amdgcn_wave_barrier();
    asm volatile("s_wait_dscnt 0" ::: "memory");

    const int idx_t = sidx[wave * 16 + m];   // nearest code for row m

    // ---- Gather codebook row, write quant, accumulate (q - x)^2.
    // Reuses the resident A fragments: x never re-read from HBM.
    float err = 0.f;
    #pragma unroll 8
    for (int kk = 0; kk < 32; ++kk) {
        const v2f q = *(const v2f*)(ws + (size_t)idx_t * SUB + kk * 4 + 2 * hi);
        const size_t a0 = xbase + (size_t)(kk * 4 + 2 * hi) * T_DIM;
        out[a0]         = q.x;
        out[a0 + T_DIM] = q.y;
        const float e0 = q.x - af[kk].x;
        const float e1 = q.y - af[kk].y;
        err += e0 * e0 + e1 * e1;
    }
    #pragma unroll
    for (int mask = 1; mask < 32; mask <<= 1) err += __shfl_xor(err, mask, 32);
    if (lane == 0) atomicAdd(lossAcc, err);
}

__global__ __launch_bounds__(512) void vq_finalize_kernel(
    const float* __restrict__ hist,
    const float* __restrict__ lossAcc,
    float* __restrict__ outScalars)   // [loss, perp]
{
    __shared__ float red[KCODES];
    const int k = threadIdx.x;
    float perp = 0.f;
    for (int s = 0; s < 4; ++s) {
        const float p = hist[s * KCODES + k] * (1.0f / (float)N_ROWS);
        red[k] = p * __logf(p + 1e-10f);
        __syncthreads();
        for (int off = KCODES / 2; off > 0; off >>= 1) {
            if (k < off) red[k] += red[k + off];
            __syncthreads();
        }
        if (k == 0) perp += __expf(-red[0]);
        __syncthreads();
    }
    if (k == 0) {
        outScalars[0] = (1.0f + BETA) * lossAcc[0] * (1.0f / (float)N_ELEM);
        outScalars[1] = perp;
    }
}

extern "C" void kernel_launch(void* const* d_in, const int* in_sizes, int n_in,
                              void* d_out, int out_size, void* d_ws, size_t ws_size,
                              hipStream_t stream) {
    const float* x  = (const float*)d_in[0];
    const float* w1 = (const float*)d_in[1];
    const float* w2 = (const float*)d_in[2];
    const float* w3 = (const float*)d_in[3];
    const float* w4 = (const float*)d_in[4];
    float* out = (float*)d_out;

    float* hist    = (float*)d_ws;          // 4*512 floats
    float* lossAcc = hist + 4 * KCODES;     // 1 float

    // Zero accumulators every launch (graph-capture safe, replay-deterministic).
    hipMemsetAsync(d_ws, 0, (4 * KCODES + 1) * sizeof(float), stream);

    // 8192 wave-tiles (4 subs x 2048 row-tiles), 8 waves per 256-thread block;
    // all waves of a block share one sub-codebook -> shared LDS staging.
    vq_main_kernel<<<1024, 256, 0, stream>>>(x, w1, w2, w3, w4, out, hist, lossAcc);
    vq_finalize_kernel<<<1, 512, 0, stream>>>(hist, lossAcc,
                                              out + (size_t)N_ELEM);
}